// PACBertEmbeddings_39556648796469
// MI455X (gfx1250) — compile-verified
//
#include <hip/hip_runtime.h>
#include <hip/hip_bf16.h>

// Problem constants (match reference)
#define Bc 64
#define Tc 512
#define Gc 64
#define Hc 768
#define GHc 256
#define Lc (Tc + Gc + 3)   // 579

typedef float v2f __attribute__((ext_vector_type(2)));
typedef float v8f __attribute__((ext_vector_type(8)));

// ---------------------------------------------------------------------------
// Kernel 0: per-batch counts  (text_cnt = #nonzero text tokens after CLS,
//                              tag_cnt  = #nonzero tag tokens)
// ---------------------------------------------------------------------------
__global__ __launch_bounds__(256)
void counts_kernel(const int* __restrict__ text_ids,
                   const int* __restrict__ tag_ids,
                   int* __restrict__ cnts) {
  int b = blockIdx.x;
  int t = threadIdx.x;
  int tc = 0;
  for (int i = 1 + t; i < Tc; i += 256) tc += (text_ids[b * Tc + i] > 0) ? 1 : 0;
  int gc = 0;
  if (t < Gc) gc = (tag_ids[b * Gc + t] > 0) ? 1 : 0;
  // wave32 reduction
  #pragma unroll
  for (int off = 16; off > 0; off >>= 1) {
    tc += __shfl_xor(tc, off);
    gc += __shfl_xor(gc, off);
  }
  __shared__ int stc[8], sgc[8];
  int wv = t >> 5, ln = t & 31;
  if (ln == 0) { stc[wv] = tc; sgc[wv] = gc; }
  __syncthreads();
  if (t == 0) {
    int a = 0, c = 0;
    #pragma unroll
    for (int i = 0; i < 8; ++i) { a += stc[i]; c += sgc[i]; }
    cnts[2 * b]     = a;
    cnts[2 * b + 1] = c;
  }
}

// ---------------------------------------------------------------------------
// Kernel 1: fused GEMM via V_WMMA_F32_16X16X4_F32
//   rows [0, B*G)      : gnn_tag_emb[tag_ids[row]] @ Wt + bt, * bert_tag_emb[tok]
//   rows [B*G, B*G+B)  : user_emb[user_ids[row-B*G]] @ Wu + bu
// Grid: (260, 12); block = 128 (4 waves); each wave -> one 16x16 output tile.
// ---------------------------------------------------------------------------
__global__ __launch_bounds__(128)
void fused_gemm(const int* __restrict__ tag_ids, const int* __restrict__ user_ids,
                const float* __restrict__ gnn_tag_emb, const float* __restrict__ user_emb,
                const float* __restrict__ Wt, const float* __restrict__ bt,
                const float* __restrict__ Wu, const float* __restrict__ bu,
                const float* __restrict__ bert_tag_emb,
                float* __restrict__ tag_e, float* __restrict__ user_e) {
  constexpr int LDA = GHc + 4;          // 260-float row stride: no LDS bank conflicts
  __shared__ float As[16 * LDA];

  const int  mTile  = blockIdx.x;                 // 0..259
  const int  nBase  = blockIdx.y * 64;            // 0..704
  const bool isUser = (mTile >= (Bc * Gc) / 16);  // tiles 256..259 are user rows

  // Stage 16 A rows (GH floats each) into LDS, float4-vectorized gather.
  for (int i = threadIdx.x; i < 16 * (GHc / 4); i += 128) {
    int r  = i / (GHc / 4);
    int c4 = i % (GHc / 4);
    int rowIdx = mTile * 16 + r;
    const float* src;
    if (!isUser) src = gnn_tag_emb + (size_t)tag_ids[rowIdx] * GHc;
    else         src = user_emb    + (size_t)user_ids[rowIdx - Bc * Gc] * GHc;
    float4 v = ((const float4*)src)[c4];
    float* dst = &As[r * LDA + c4 * 4];
    dst[0] = v.x; dst[1] = v.y; dst[2] = v.z; dst[3] = v.w;
  }
  __syncthreads();

  const int wave = threadIdx.x >> 5;
  const int lane = threadIdx.x & 31;
  const int m    = lane & 15;   // A row within tile / C column index source
  const int kh   = lane >> 4;   // k-half select
  const int ncol = nBase + wave * 16 + (lane & 15);

  const float* __restrict__ W = isUser ? Wu : Wt;
  const float* arow = &As[m * LDA + 2 * kh];
  const float* wcol = W + (size_t)(2 * kh) * Hc + ncol;

  v8f acc = {0.f, 0.f, 0.f, 0.f, 0.f, 0.f, 0.f, 0.f};
  #pragma unroll 4
  for (int k0 = 0; k0 < GHc; k0 += 4) {
    v2f a, b;
    a.x = arow[k0];
    a.y = arow[k0 + 1];
    b.x = wcol[(size_t)k0 * Hc];
    b.y = wcol[(size_t)(k0 + 1) * Hc];
    // D = A(16x4) * B(4x16) + C ; f32 in, f32 accumulate (matches reference fp32)
    acc = __builtin_amdgcn_wmma_f32_16x16x4_f32(
        /*neg_a=*/false, a, /*neg_b=*/false, b,
        /*c_mod=*/(short)0, acc, /*reuse_a=*/false, /*reuse_b=*/false);
  }

  const float bias = (isUser ? bu : bt)[ncol];
  #pragma unroll
  for (int j = 0; j < 8; ++j) {
    int mrow   = 8 * kh + j;                 // C/D layout: VGPR j, half kh -> row
    int rowIdx = mTile * 16 + mrow;
    float v = acc[j] + bias;
    if (!isUser) {
      int tok = tag_ids[rowIdx];
      v *= bert_tag_emb[(size_t)tok * Hc + ncol];
      tag_e[(size_t)rowIdx * Hc + ncol] = v;
    } else {
      user_e[(size_t)(rowIdx - Bc * Gc) * Hc + ncol] = v;
    }
  }
}

// ---------------------------------------------------------------------------
// Kernel 2: assemble sequence rows + type/pos embeddings + LayerNorm + mask.
// One wave (32 lanes) per (b, p) row; 24 floats per lane (H = 768).
// ---------------------------------------------------------------------------
__global__ __launch_bounds__(256)
void assemble_ln(const int* __restrict__ text_ids, const int* __restrict__ cnts,
                 const float* __restrict__ word_emb, const float* __restrict__ pos_emb,
                 const float* __restrict__ type_emb,
                 const float* __restrict__ ln_g, const float* __restrict__ ln_b,
                 const float* __restrict__ tag_e, const float* __restrict__ user_e,
                 float* __restrict__ out_emb, float* __restrict__ out_mask) {
  const int wave = threadIdx.x >> 5;
  const int lane = threadIdx.x & 31;
  const int row  = blockIdx.x * 8 + wave;
  if (row >= Bc * Lc) return;
  const int b = row / Lc;
  const int p = row % Lc;

  const int text_cnt = cnts[2 * b];
  const int tag_cnt  = cnts[2 * b + 1];
  const int text_end = 3 + text_cnt;
  const int tag_end  = text_end + tag_cnt;

  // Source row selection (matches reference where-chain precedence).
  const float* base = nullptr;
  if (p == 0)                                base = word_emb + (size_t)text_ids[b * Tc] * Hc;
  else if (p == 1)                           base = user_e + (size_t)b * Hc;
  else if (p == 2 || p == tag_end)           base = word_emb + (size_t)text_ids[b * Tc + text_cnt] * Hc;
  else if (p >= 3 && p < text_end)           base = word_emb + (size_t)text_ids[b * Tc + (p - 2)] * Hc;
  else if (p >= text_end && p < tag_end)     base = tag_e + (size_t)(b * Gc + (p - text_end)) * Hc;

  const int tt  = (p >= 3 && p < text_end) ? 1
                : ((p >= text_end && p <= tag_end) ? 2 : 0);
  const int pos = (p >= text_end && p < tag_end) ? text_end
                : ((p >= tag_end) ? (p - tag_cnt + 1) : p);

  const float* te = type_emb + (size_t)tt * Hc;
  const float* pe = pos_emb + (size_t)pos * Hc;

  float v[Hc / 32];
  float s = 0.f, s2 = 0.f;
  #pragma unroll
  for (int j = 0; j < Hc / 32; ++j) {
    int idx = lane + 32 * j;
    float x = (base ? base[idx] : 0.f) + te[idx] + pe[idx];
    v[j] = x;
    s  += x;
    s2 += x * x;
  }
  #pragma unroll
  for (int off = 16; off > 0; off >>= 1) {
    s  += __shfl_xor(s, off);
    s2 += __shfl_xor(s2, off);
  }
  const float mu  = s * (1.f / Hc);
  const float var = s2 * (1.f / Hc) - mu * mu;
  const float rs  = rsqrtf(var + 1e-12f);

  float* o = out_emb + (size_t)row * Hc;
  #pragma unroll
  for (int j = 0; j < Hc / 32; ++j) {
    int idx = lane + 32 * j;
    o[idx] = ln_g[idx] * (v[j] - mu) * rs + ln_b[idx];
  }
  if (lane == 0) out_mask[row] = (p <= tag_end) ? 1.0f : 0.0f;
}

// ---------------------------------------------------------------------------
extern "C" void kernel_launch(void* const* d_in, const int* in_sizes, int n_in,
                              void* d_out, int out_size, void* d_ws, size_t ws_size,
                              hipStream_t stream) {
  const int*   user_ids     = (const int*)d_in[0];
  const int*   text_ids     = (const int*)d_in[1];
  const int*   tag_ids      = (const int*)d_in[2];
  const float* user_emb     = (const float*)d_in[3];
  const float* word_emb     = (const float*)d_in[4];
  const float* bert_tag_emb = (const float*)d_in[5];
  const float* gnn_tag_emb  = (const float*)d_in[6];
  const float* pos_emb      = (const float*)d_in[7];
  const float* type_emb     = (const float*)d_in[8];
  const float* Wu           = (const float*)d_in[9];
  const float* bu           = (const float*)d_in[10];
  const float* Wt           = (const float*)d_in[11];
  const float* bt           = (const float*)d_in[12];
  const float* ln_g         = (const float*)d_in[13];
  const float* ln_b         = (const float*)d_in[14];

  char*  ws     = (char*)d_ws;
  int*   cnts   = (int*)ws;                                   // 2*B ints
  float* user_e = (float*)(ws + 4096);                        // B*H floats
  float* tag_e  = (float*)(ws + 4096 + (size_t)Bc * Hc * 4);  // B*G*H floats

  float* out_emb  = (float*)d_out;
  float* out_mask = out_emb + (size_t)Bc * Lc * Hc;

  counts_kernel<<<Bc, 256, 0, stream>>>(text_ids, tag_ids, cnts);

  dim3 g1((Bc * Gc + Bc) / 16, Hc / 64);  // (260, 12)
  fused_gemm<<<g1, 128, 0, stream>>>(tag_ids, user_ids, gnn_tag_emb, user_emb,
                                     Wt, bt, Wu, bu, bert_tag_emb, tag_e, user_e);

  const int nrows = Bc * Lc;               // 37056
  assemble_ln<<<(nrows + 7) / 8, 256, 0, stream>>>(
      text_ids, cnts, word_emb, pos_emb, type_emb, ln_g, ln_b,
      tag_e, user_e, out_emb, out_mask);
}